// Net_28389733826738
// MI455X (gfx1250) — compile-verified
//
#include <hip/hip_runtime.h>
#include <hip/hip_bf16.h>
#include <math.h>

// ---------------------------------------------------------------------------
// 2-layer GCN on MI455X (gfx1250, wave32).
//  - GEMMs use native fp32 WMMA: V_WMMA_F32_16X16X4_F32 (one 16x16 tile/wave).
//    GEMM1 is HBM-bound on reading x (102 MB ~ 4.4us at 23.3 TB/s); fp32 WMMA
//    keeps full reference precision since compute is nowhere near the limit.
//  - Edge aggregation: thread-per-(edge,feature) f32 atomics; accumulators
//    (N*16 = 6.4MB, N*40 = 16MB) are L2-resident (192MB L2), so atomics
//    resolve in L2's atomic units — the workload's real bottleneck.
// ---------------------------------------------------------------------------

typedef __attribute__((ext_vector_type(2))) float v2f;
typedef __attribute__((ext_vector_type(8))) float v8f;

#define F_IN 256
#define HID  16
#define NC   40

// ---- degree / normalization ----------------------------------------------

__global__ void k_deg_init(float* __restrict__ deg, int n) {
  int i = blockIdx.x * blockDim.x + threadIdx.x;
  if (i < n) deg[i] = 1.0f;  // self-loop contributes 1 to every node degree
}

__global__ void k_deg_accum(const int* __restrict__ dst, float* __restrict__ deg, int e) {
  int i = blockIdx.x * blockDim.x + threadIdx.x;
  if (i < e) atomicAdd(&deg[dst[i]], 1.0f);
}

__global__ void k_dinv(float* __restrict__ deg, int n) {
  int i = blockIdx.x * blockDim.x + threadIdx.x;
  if (i < n) deg[i] = rsqrtf(deg[i]);  // deg >= 1 always (self loops)
}

// ---- GEMM1: h1[N x 16] = x[N x 256] @ W1[256 x 16], fp32 WMMA -------------
// One wave owns one 16-row x 16-col tile; K loop of 64 x (16x16x4) WMMAs.
// A layout (16x4 f32): lanes 0-15 -> M=lane, K = k0+{0,1}; lanes 16-31 ->
// M=lane-16, K = k0+{2,3}. B layout mirrors (rows striped across lanes).
// C/D: VGPR r -> row r (lanes 0-15) / row r+8 (lanes 16-31), col = lane&15.

__global__ void k_gemm1(const float* __restrict__ x, const float* __restrict__ W1,
                        float* __restrict__ h1, int n) {
  int gwave = (int)((blockIdx.x * (long long)blockDim.x + threadIdx.x) >> 5);
  int lane  = threadIdx.x & 31;
  int row0  = gwave * 16;
  if (row0 >= n) return;

  int m  = lane & 15;
  int kh = (lane >> 4) * 2;        // 0 for lanes 0-15, 2 for lanes 16-31
  v8f acc = {};
  for (int k0 = 0; k0 < F_IN; k0 += 4) {
    const float* ap = x + (size_t)(row0 + m) * F_IN + (k0 + kh);
    v2f a = {ap[0], ap[1]};
    const float* bp = W1 + (size_t)(k0 + kh) * HID + m;
    v2f b = {bp[0], bp[HID]};
    acc = __builtin_amdgcn_wmma_f32_16x16x4_f32(
        /*neg_a=*/false, a, /*neg_b=*/false, b,
        /*c_mod=*/(short)0, acc, /*reuse_a=*/false, /*reuse_b=*/false);
  }
  int mb = row0 + ((lane >> 4) ? 8 : 0);
  int c  = lane & 15;
  if (row0 + 16 <= n) {
    // full tile (always the case for N % 16 == 0): unguarded coalesced stores
#pragma unroll
    for (int r = 0; r < 8; ++r) h1[(size_t)(mb + r) * HID + c] = acc[r];
  } else {
#pragma unroll
    for (int r = 0; r < 8; ++r) {
      int row = mb + r;
      if (row < n) h1[(size_t)row * HID + c] = acc[r];
    }
  }
}

// ---- self-loop init: out[i][j] = dinv[i]^2 * h[i][j]  (fully writes out) --

__global__ void k_selfloop16(const float* __restrict__ h, const float* __restrict__ dinv,
                             float* __restrict__ out, long long total) {
  long long t = blockIdx.x * (long long)blockDim.x + threadIdx.x;
  if (t >= total) return;
  int i = (int)(t >> 4);
  float d = dinv[i];
  out[t] = d * d * h[t];
}

__global__ void k_selfloop40(const float* __restrict__ h, const float* __restrict__ dinv,
                             float* __restrict__ out, long long total) {
  long long t = blockIdx.x * (long long)blockDim.x + threadIdx.x;
  if (t >= total) return;
  int i = (int)(t / NC);
  float d = dinv[i];
  out[t] = d * d * h[t];
}

// ---- edge scatter: out[dst] += dinv[src]*dinv[dst] * h[src] ---------------

__global__ void k_scatter16(const int* __restrict__ src, const int* __restrict__ dst,
                            const float* __restrict__ dinv, const float* __restrict__ h,
                            float* __restrict__ out, long long total) {
  long long t = blockIdx.x * (long long)blockDim.x + threadIdx.x;
  if (t >= total) return;
  int e = (int)(t >> 4);
  int j = (int)(t & 15);
  int s = src[e], d = dst[e];
  float nrm = dinv[s] * dinv[d];
  atomicAdd(&out[(size_t)d * HID + j], nrm * h[(size_t)s * HID + j]);
}

__global__ void k_scatter40(const int* __restrict__ src, const int* __restrict__ dst,
                            const float* __restrict__ dinv, const float* __restrict__ h,
                            float* __restrict__ out, long long total) {
  long long t = blockIdx.x * (long long)blockDim.x + threadIdx.x;
  if (t >= total) return;
  int e = (int)(t / NC);
  int j = (int)(t - (long long)e * NC);
  int s = src[e], d = dst[e];
  float nrm = dinv[s] * dinv[d];
  atomicAdd(&out[(size_t)d * NC + j], nrm * h[(size_t)s * NC + j]);
}

// ---- bias + ReLU (in place, layer 1) --------------------------------------

__global__ void k_bias_relu(float* __restrict__ a, const float* __restrict__ b1,
                            long long total) {
  long long t = blockIdx.x * (long long)blockDim.x + threadIdx.x;
  if (t >= total) return;
  float v = a[t] + b1[(int)(t & 15)];
  a[t] = v > 0.0f ? v : 0.0f;
}

// ---- GEMM2: p2[N x 40] = a1[N x 16] @ W2[16 x 40], fp32 WMMA --------------
// 3 column tiles of 16 (40 padded to 48); B loads/stores bounds-checked on
// the column only; full-row fast path avoids per-row exec-mask churn.

__global__ void k_gemm2(const float* __restrict__ a1, const float* __restrict__ W2,
                        float* __restrict__ p2, int n) {
  int gwave = (int)((blockIdx.x * (long long)blockDim.x + threadIdx.x) >> 5);
  int lane  = threadIdx.x & 31;
  int ct    = gwave % 3;
  int rt    = gwave / 3;
  int row0  = rt * 16;
  if (row0 >= n) return;

  int m  = lane & 15;
  int kh = (lane >> 4) * 2;
  int cb = ct * 16;
  v8f acc = {};
#pragma unroll
  for (int k0 = 0; k0 < HID; k0 += 4) {
    const float* ap = a1 + (size_t)(row0 + m) * HID + (k0 + kh);
    v2f a = {ap[0], ap[1]};
    v2f b = {0.0f, 0.0f};
    int c = cb + m;
    if (c < NC) {
      const float* bp = W2 + (size_t)(k0 + kh) * NC + c;
      b[0] = bp[0];
      b[1] = bp[NC];
    }
    acc = __builtin_amdgcn_wmma_f32_16x16x4_f32(
        false, a, false, b, (short)0, acc, false, false);
  }
  int mb = row0 + ((lane >> 4) ? 8 : 0);
  int c  = cb + (lane & 15);
  if (c < NC) {
    if (row0 + 16 <= n) {
#pragma unroll
      for (int r = 0; r < 8; ++r) p2[(size_t)(mb + r) * NC + c] = acc[r];
    } else {
#pragma unroll
      for (int r = 0; r < 8; ++r) {
        int row = mb + r;
        if (row < n) p2[(size_t)row * NC + c] = acc[r];
      }
    }
  }
}

// ---- bias + log_softmax over 40 classes (one thread per node) -------------

__global__ void k_logsoftmax(float* __restrict__ out, const float* __restrict__ b2, int n) {
  int i = blockIdx.x * blockDim.x + threadIdx.x;
  if (i >= n) return;
  float v[NC];
  float mx = -INFINITY;
#pragma unroll
  for (int c = 0; c < NC; ++c) {
    v[c] = out[(size_t)i * NC + c] + b2[c];
    mx = fmaxf(mx, v[c]);
  }
  float s = 0.0f;
#pragma unroll
  for (int c = 0; c < NC; ++c) s += __expf(v[c] - mx);
  float ls = __logf(s);
#pragma unroll
  for (int c = 0; c < NC; ++c) out[(size_t)i * NC + c] = v[c] - mx - ls;
}

// ---------------------------------------------------------------------------

extern "C" void kernel_launch(void* const* d_in, const int* in_sizes, int n_in,
                              void* d_out, int out_size, void* d_ws, size_t ws_size,
                              hipStream_t stream) {
  const float* x  = (const float*)d_in[0];
  const int*   ei = (const int*)d_in[1];   // [2, E] flat: src then dst
  const float* W1 = (const float*)d_in[2];
  const float* b1 = (const float*)d_in[3];
  const float* W2 = (const float*)d_in[4];
  const float* b2 = (const float*)d_in[5];
  float* out = (float*)d_out;              // [N, 40]

  const int n = in_sizes[0] / F_IN;
  const int e = in_sizes[1] / 2;
  const int* src = ei;
  const int* dst = ei + e;

  // workspace layout (floats): dinv[n] | h1[n*16] | a1[n*16] | p2[n*40]
  float* ws   = (float*)d_ws;
  float* dinv = ws;
  float* h1   = dinv + n;
  float* a1   = h1 + (size_t)n * HID;
  float* p2   = a1 + (size_t)n * HID;

  const int B = 256;
  const int nb_n = (n + B - 1) / B;

  // degrees -> dinv
  k_deg_init <<<nb_n, B, 0, stream>>>(dinv, n);
  k_deg_accum<<<(e + B - 1) / B, B, 0, stream>>>(dst, dinv, e);
  k_dinv     <<<nb_n, B, 0, stream>>>(dinv, n);

  // layer 1: transform, self-loop init, edge scatter, bias+relu
  {
    int waves = (n + 15) / 16;
    int thr   = waves * 32;
    k_gemm1<<<(thr + B - 1) / B, B, 0, stream>>>(x, W1, h1, n);
  }
  {
    long long tot = (long long)n * HID;
    k_selfloop16<<<(int)((tot + B - 1) / B), B, 0, stream>>>(h1, dinv, a1, tot);
    long long etot = (long long)e * HID;
    k_scatter16 <<<(int)((etot + B - 1) / B), B, 0, stream>>>(src, dst, dinv, h1, a1, etot);
    k_bias_relu <<<(int)((tot + B - 1) / B), B, 0, stream>>>(a1, b1, tot);
  }

  // layer 2: transform, self-loop init (into d_out), edge scatter, bias+log_softmax
  {
    int waves = ((n + 15) / 16) * 3;  // 3 column tiles of 16 (40 -> 48)
    int thr   = waves * 32;
    k_gemm2<<<(thr + B - 1) / B, B, 0, stream>>>(a1, W2, p2, n);
  }
  {
    long long tot = (long long)n * NC;
    k_selfloop40<<<(int)((tot + B - 1) / B), B, 0, stream>>>(p2, dinv, out, tot);
    long long etot = (long long)e * NC;
    k_scatter40 <<<(int)((etot + B - 1) / B), B, 0, stream>>>(src, dst, dinv, p2, out, etot);
  }
  k_logsoftmax<<<nb_n, B, 0, stream>>>(out, b2, n);
}